// PermutationPlackettLayer_56762287784057
// MI455X (gfx1250) — compile-verified
//
#include <hip/hip_runtime.h>
#include <hip/hip_bf16.h>
#include <math.h>

typedef float v2f __attribute__((ext_vector_type(2)));
typedef float v4f __attribute__((ext_vector_type(4)));
typedef float v8f __attribute__((ext_vector_type(8)));

#define B_   16
#define AR_  2
#define BA_  32
#define N_   1024
#define H_   1024
#define F_   512

// ---------------------------------------------------------------------------
// Kernel 1: fused MLP  logits = relu(q @ W1 + b1) @ W2 + b2 ;
//           acti = log(max(softplus(logits),1e-5)) + uniform_noise
// One WG = 16 rows of q. 8 waves; wave w owns hidden cols [64w, 64w+64).
// f32 WMMA 16x16x4, A staged in LDS (two 32KB chunks), B streamed from L2.
// ---------------------------------------------------------------------------
__global__ __launch_bounds__(256) void mlp_acti_kernel(
    const float* __restrict__ q,    // [16*1024, 1024]
    const float* __restrict__ W1,   // [1024, 512]
    const float* __restrict__ b1,   // [512]
    const float* __restrict__ W2,   // [512]
    const float* __restrict__ b2,   // [1]
    const float* __restrict__ un,   // [16*1024]
    float* __restrict__ ws_acti)    // [16*1024]
{
    __shared__ float lA[16 * 512];      // 32 KB chunk of A (16 rows x 512 K)
    __shared__ float accw[8 * 16];      // per-wave partial logits

    const int tile  = blockIdx.x;       // 0..1023 (16 rows each)
    const int row0  = tile * 16;
    const int tid   = threadIdx.x;
    const int wave  = tid >> 5;
    const int lane  = tid & 31;
    const int m     = lane & 15;        // A row / B col within 16
    const int khalf = lane >> 4;        // K pair selector (0 -> K0/K1, 1 -> K2/K3)
    const int colbase = wave * 64;

    v8f c0 = {}, c1 = {}, c2 = {}, c3 = {};

    const v4f* qv  = (const v4f*)q;
    v4f*       lAv = (v4f*)lA;

    for (int kc = 0; kc < 2; ++kc) {
        __syncthreads();                 // previous chunk fully consumed
        for (int i = tid; i < 16 * 128; i += 256) {
            int r  = i >> 7;             // row 0..15
            int cv = i & 127;            // float4 col within chunk
            lAv[i] = qv[(size_t)(row0 + r) * 256 + (size_t)kc * 128 + cv];
        }
        __syncthreads();

        for (int k0 = 0; k0 < 512; k0 += 4) {
            int kk = k0 + khalf * 2;                       // K within chunk
            v2f a = *(const v2f*)&lA[m * 512 + kk];        // A: rows m, K kk..kk+1
            int kg = kc * 512 + kk;                        // global K
            const float* w1r0 = W1 + (size_t)kg * F_;
            const float* w1r1 = w1r0 + F_;
            int n0 = colbase + m;
            v2f bv0 = { w1r0[n0],      w1r1[n0]      };
            v2f bv1 = { w1r0[n0 + 16], w1r1[n0 + 16] };
            v2f bv2 = { w1r0[n0 + 32], w1r1[n0 + 32] };
            v2f bv3 = { w1r0[n0 + 48], w1r1[n0 + 48] };
            c0 = __builtin_amdgcn_wmma_f32_16x16x4_f32(false, a, false, bv0, (short)0, c0, false, false);
            c1 = __builtin_amdgcn_wmma_f32_16x16x4_f32(false, a, false, bv1, (short)0, c1, false, false);
            c2 = __builtin_amdgcn_wmma_f32_16x16x4_f32(false, a, false, bv2, (short)0, c2, false, false);
            c3 = __builtin_amdgcn_wmma_f32_16x16x4_f32(false, a, false, bv3, (short)0, c3, false, false);
        }
    }

    // Epilogue: h = relu(c + b1); rowsum[r] += h * W2  (rows r + 8*khalf)
    float rowsum[8];
#pragma unroll
    for (int r = 0; r < 8; ++r) rowsum[r] = 0.f;
    const int nC = lane & 15;
    v8f cs[4] = { c0, c1, c2, c3 };
#pragma unroll
    for (int t = 0; t < 4; ++t) {
        int col = colbase + t * 16 + nC;
        float bb = b1[col];
        float ww = W2[col];
#pragma unroll
        for (int r = 0; r < 8; ++r) {
            float h = cs[t][r] + bb;
            h = fmaxf(h, 0.f);
            rowsum[r] = fmaf(h, ww, rowsum[r]);
        }
    }
    // butterfly sum across the 16 lanes sharing khalf (offsets 1..8 stay in group)
#pragma unroll
    for (int off = 1; off < 16; off <<= 1) {
#pragma unroll
        for (int r = 0; r < 8; ++r)
            rowsum[r] += __shfl_xor(rowsum[r], off, 32);
    }
    if (nC == 0) {
#pragma unroll
        for (int r = 0; r < 8; ++r)
            accw[wave * 16 + khalf * 8 + r] = rowsum[r];
    }
    __syncthreads();

    if (tid < 16) {
        float logit = b2[0];
#pragma unroll
        for (int w = 0; w < 8; ++w) logit += accw[w * 16 + tid];
        // stable softplus
        float sp = fmaxf(logit, 0.f) + log1pf(expf(-fabsf(logit)));
        float a  = logf(fmaxf(sp, 1e-5f)) + un[row0 + tid];
        ws_acti[row0 + tid] = a;
    }
}

// ---------------------------------------------------------------------------
// Kernel 2a: per Ba-batch, compute masked_sqz (output), s_j, and
//            base_j = -Bm_j - (1-mask_j)*1e9  (Bm_j = sum_k |s_j - s_k|)
// ---------------------------------------------------------------------------
__global__ __launch_bounds__(1024) void prep_kernel(
    const float* __restrict__ ws_acti,   // [16*1024]
    const float* __restrict__ qmask,     // [16*1024]
    const float* __restrict__ gumbel,    // [32*1024]
    float* __restrict__ out_masked,      // [32*1024]
    float* __restrict__ ws_s,            // [32*1024]
    float* __restrict__ ws_base)         // [32*1024]
{
    __shared__ float ls[N_];
    int ba = blockIdx.x, j = threadIdx.x;
    int src = ba / AR_;
    float mk = qmask[src * N_ + j];
    float av = ws_acti[src * N_ + j];
    float masked = (mk > 0.f) ? av : -1e6f;
    out_masked[(size_t)ba * N_ + j] = masked;
    float pert = masked + gumbel[ba * N_ + j];
    float s = expf(pert) * mk;
    ls[j] = s;
    ws_s[ba * N_ + j] = s;
    __syncthreads();
    float bm = 0.f;
    for (int k = 0; k < N_; ++k) bm += fabsf(s - ls[k]);
    ws_base[ba * N_ + j] = -bm - (1.f - mk) * 1e9f;
}

// ---------------------------------------------------------------------------
// Kernel 2b: greedy sequential argmax (find_permu). One wave per batch.
// Row i values = s_j * (1023-2i) + base_j - flag_j*1e9. All state in VGPRs:
// 32 (s,base) pairs + 32-bit flag mask per lane; barrier-free shuffle argmax.
// First-index tie-break preserved.
// ---------------------------------------------------------------------------
__global__ __launch_bounds__(32) void greedy_kernel(
    const float* __restrict__ ws_s,
    const float* __restrict__ ws_base,
    int* __restrict__ ws_chosen)         // [32*1024]
{
    int ba = blockIdx.x;
    int lane = threadIdx.x;
    const float* S  = ws_s   + (size_t)ba * N_;
    const float* Bb = ws_base + (size_t)ba * N_;
    float sv[32], bv[32];
#pragma unroll
    for (int e = 0; e < 32; ++e) {
        int j = e * 32 + lane;           // coalesced, j = (e<<5)|lane
        sv[e] = S[j];
        bv[e] = Bb[j];
    }
    unsigned flag = 0u;
    for (int i = 0; i < N_; ++i) {
        float scale = (float)(N_ - 1 - 2 * i);     // 1023 - 2i
        float best = -3.4e38f;
        int bidx = 0x7fffffff;
#pragma unroll
        for (int e = 0; e < 32; ++e) {
            float v = fmaf(sv[e], scale, bv[e]);
            if (flag & (1u << e)) v -= 1e9f;
            int j = e * 32 + lane;
            if (v > best) { best = v; bidx = j; }  // e ascending => keeps min j on ties
        }
#pragma unroll
        for (int off = 16; off > 0; off >>= 1) {
            float ov = __shfl_xor(best, off, 32);
            int   oj = __shfl_xor(bidx, off, 32);
            if (ov > best || (ov == best && oj < bidx)) { best = ov; bidx = oj; }
        }
        if ((bidx & 31) == lane) flag |= 1u << (bidx >> 5);
        if (lane == 0) ws_chosen[(size_t)ba * N_ + i] = bidx;
    }
}

// ---------------------------------------------------------------------------
// Kernel 3a: permu = where(mask>0, shift_right(chosen, fill 0), identity)
// ---------------------------------------------------------------------------
__global__ __launch_bounds__(1024) void permu_kernel(
    const float* __restrict__ qmask,
    const int* __restrict__ ws_chosen,
    int* __restrict__ ws_permu)
{
    int ba = blockIdx.x, i = threadIdx.x;
    int src = ba / AR_;
    float mi = qmask[src * N_ + i];
    int ch = (i == 0) ? 0 : ws_chosen[(size_t)ba * N_ + i - 1];
    ws_permu[(size_t)ba * N_ + i] = (mi > 0.f) ? ch : i;
}

// ---------------------------------------------------------------------------
// Kernel 3b: sample_permu one-hot, 128 MB streaming write (NT ext-vec stores)
// One block per (ba, i) row.
// ---------------------------------------------------------------------------
__global__ __launch_bounds__(256) void onehot_kernel(
    const int* __restrict__ ws_permu,
    float* __restrict__ out)             // [32*1024, 1024]
{
    size_t row = blockIdx.x;
    int p = ws_permu[row];
    int c = threadIdx.x * 4;
    v4f v;
    v.x = (p == c    ) ? 1.f : 0.f;
    v.y = (p == c + 1) ? 1.f : 0.f;
    v.z = (p == c + 2) ? 1.f : 0.f;
    v.w = (p == c + 3) ? 1.f : 0.f;
    __builtin_nontemporal_store(v, (v4f*)(out + row * N_) + threadIdx.x);
}

// ---------------------------------------------------------------------------
// Kernel 3c: gather exp(masked[permu]), suffix cumsum, log sums, kl.
// Deterministic reductions (shuffle + per-wave LDS slots, no float atomics).
// ---------------------------------------------------------------------------
__global__ __launch_bounds__(1024) void stats_kernel(
    const float* __restrict__ qmask,
    const float* __restrict__ ws_acti,
    const int* __restrict__ ws_permu,
    const float* __restrict__ out_masked,
    float* __restrict__ out_kl,
    float* __restrict__ out_lognom,
    float* __restrict__ out_lognorm)
{
    __shared__ float buf[N_];
    __shared__ float red[32 * 5];
    int ba = blockIdx.x, i = threadIdx.x, src = ba / AR_;
    float mi = qmask[src * N_ + i];
    int p = ws_permu[(size_t)ba * N_ + i];
    float mp = out_masked[(size_t)ba * N_ + p];
    float ea = expf(mp);
    buf[i] = ea;
    __syncthreads();
    // inclusive suffix scan (reverse cumsum), Hillis-Steele
    for (int off = 1; off < N_; off <<= 1) {
        float add = (i + off < N_) ? buf[i + off] : 0.f;
        __syncthreads();
        buf[i] += add;
        __syncthreads();
    }
    float rc = buf[i];

    float v_nv   = mi;
    float v_lnom = logf(ea + 1e-20f) * mi;
    float v_lnor = logf(rc + 1e-20f) * mi;
    float sa = ws_acti[src * N_ + i];       // sqz_acti (pre-mask, with noise)
    float sm = sa * mi;
    float v_s1 = sm;                         // TEMP = 1
    float v_s2 = expf(-fmaxf(sm, -20.f));

    int lane = i & 31, wave = i >> 5;
#pragma unroll
    for (int off = 16; off > 0; off >>= 1) {
        v_nv   += __shfl_xor(v_nv,   off, 32);
        v_lnom += __shfl_xor(v_lnom, off, 32);
        v_lnor += __shfl_xor(v_lnor, off, 32);
        v_s1   += __shfl_xor(v_s1,   off, 32);
        v_s2   += __shfl_xor(v_s2,   off, 32);
    }
    if (lane == 0) {
        red[wave * 5 + 0] = v_nv;
        red[wave * 5 + 1] = v_lnom;
        red[wave * 5 + 2] = v_lnor;
        red[wave * 5 + 3] = v_s1;
        red[wave * 5 + 4] = v_s2;
    }
    __syncthreads();
    if (i == 0) {
        float nv = 0, lnom = 0, lnor = 0, s1 = 0, s2 = 0;
        for (int w = 0; w < 32; ++w) {
            nv   += red[w * 5 + 0];
            lnom += red[w * 5 + 1];
            lnor += red[w * 5 + 2];
            s1   += red[w * 5 + 3];
            s2   += red[w * 5 + 4];
        }
        const float TEMP = 1.0f;
        const float EG = 0.57721566490153286f;
        float kl1 = nv * (logf(TEMP) - 1.0f + EG * (1.0f / TEMP - 1.0f));
        float s2a = s2 - ((float)N_ - nv);
        float kl = kl1 + s1 / TEMP + s2a * expf(lgammaf(1.0f + 1.0f / TEMP));
        out_kl[ba]      = kl;
        out_lognom[ba]  = lnom;
        out_lognorm[ba] = lnor;
    }
}

// ---------------------------------------------------------------------------
extern "C" void kernel_launch(void* const* d_in, const int* in_sizes, int n_in,
                              void* d_out, int out_size, void* d_ws, size_t ws_size,
                              hipStream_t stream) {
    (void)in_sizes; (void)n_in; (void)out_size; (void)ws_size;
    const float* q     = (const float*)d_in[0];   // [16,1024,1024]
    const float* qmask = (const float*)d_in[1];   // [16,1024]
    const float* W1    = (const float*)d_in[2];   // [1024,512]
    const float* b1    = (const float*)d_in[3];   // [512]
    const float* W2    = (const float*)d_in[4];   // [512,1]
    const float* b2    = (const float*)d_in[5];   // [1]
    const float* un    = (const float*)d_in[6];   // [16,1024,1]
    const float* gum   = (const float*)d_in[7];   // [32,1024,1]
    // d_in[8] = action_refinement (== AR_, compile-time here)

    float* out         = (float*)d_out;
    float* out_permu   = out;                                   // 32*1024*1024
    float* out_masked  = out + (size_t)BA_ * N_ * N_;           // 32*1024
    float* out_kl      = out_masked + BA_ * N_;                 // 32
    float* out_lognom  = out_kl + BA_;                          // 32
    float* out_lognorm = out_lognom + BA_;                      // 32

    float* ws_acti  = (float*)d_ws;                             // 16*1024
    float* ws_s     = ws_acti + B_ * N_;                        // 32*1024
    float* ws_base  = ws_s + BA_ * N_;                          // 32*1024
    int*   ws_chosen = (int*)(ws_base + BA_ * N_);              // 32*1024
    int*   ws_permu  = ws_chosen + BA_ * N_;                    // 32*1024

    mlp_acti_kernel<<<1024, 256, 0, stream>>>(q, W1, b1, W2, b2, un, ws_acti);
    prep_kernel<<<BA_, 1024, 0, stream>>>(ws_acti, qmask, gum, out_masked, ws_s, ws_base);
    greedy_kernel<<<BA_, 32, 0, stream>>>(ws_s, ws_base, ws_chosen);
    permu_kernel<<<BA_, 1024, 0, stream>>>(qmask, ws_chosen, ws_permu);
    onehot_kernel<<<(unsigned)(BA_ * N_), 256, 0, stream>>>(ws_permu, out_permu);
    stats_kernel<<<BA_, 1024, 0, stream>>>(qmask, ws_acti, ws_permu, out_masked,
                                           out_kl, out_lognom, out_lognorm);
}